// LDTW_8718783610997
// MI455X (gfx1250) — compile-verified
//
#include <hip/hip_runtime.h>

#define BATCH 256
#define LSEQ  128
#define HDIM  512
#define BIGV  1e10f
#define DIAG_STRIDE 32768   // padded per-batch floats for diagonal-major dist (255*128 = 32640 used)

typedef float v2f __attribute__((ext_vector_type(2)));
typedef float v8f __attribute__((ext_vector_type(8)));

// ---------------------------------------------------------------------------
// Phase 1: batched sim = qry @ cnd^T via V_WMMA_F32_16X16X4_F32, fused with
// mask + negation, stored in DIAGONAL-major layout for the DP phase.
//   grid = (8 col-tiles, 256 batches), block = 256 threads (8 wave32);
//   wave w computes the 16x16 tile at rows [16w,16w+16), cols [16*bx,...).
// Two independent accumulator chains (even/odd K-quads) so the XDL pipeline
// always has a ready WMMA to interleave with the loads.
// ---------------------------------------------------------------------------
__global__ __launch_bounds__(256)
void sim_dist_wmma(const float* __restrict__ qry, const float* __restrict__ cnd,
                   const float* __restrict__ qmsk, const float* __restrict__ cmsk,
                   float* __restrict__ wsdiag) {
  const int b    = blockIdx.y;
  const int n0   = blockIdx.x << 4;       // tile col base
  const int wave = threadIdx.x >> 5;
  const int lane = threadIdx.x & 31;
  const int m0   = wave << 4;             // tile row base

  const float* Aq = qry + (size_t)b * LSEQ * HDIM;
  const float* Bc = cnd + (size_t)b * LSEQ * HDIM;

  // A 16x4 layout: lane holds row (lane&15), K pair ((lane>>4)*2)
  // B  4x16 layout: lane holds col (lane&15), K pair ((lane>>4)*2); B[k,n]=cnd[n,k]
  const int idx  = lane & 15;
  const int koff = (lane >> 4) << 1;
  const float* ap = Aq + (size_t)(m0 + idx) * HDIM + koff;
  const float* bp = Bc + (size_t)(n0 + idx) * HDIM + koff;

  v8f acc0 = {};
  v8f acc1 = {};
#pragma unroll 2
  for (int k = 0; k < HDIM; k += 8) {
    v2f a0 = *(const v2f*)(ap + k);
    v2f b0 = *(const v2f*)(bp + k);
    v2f a1 = *(const v2f*)(ap + k + 4);
    v2f b1 = *(const v2f*)(bp + k + 4);
    // 8 args: (neg_a, A, neg_b, B, c_mod, C, reuse_a, reuse_b)
    acc0 = __builtin_amdgcn_wmma_f32_16x16x4_f32(false, a0, false, b0,
                                                 (short)0, acc0, false, false);
    acc1 = __builtin_amdgcn_wmma_f32_16x16x4_f32(false, a1, false, b1,
                                                 (short)0, acc1, false, false);
  }
  v8f acc = acc0 + acc1;

  // Epilogue: C/D layout -> lane half g = lane>>4 holds rows v+8g, col lane&15.
  const int cc   = n0 + (lane & 15);
  const int half = lane >> 4;
  const float cmv = cmsk[b * LSEQ + cc];
  float* outb = wsdiag + (size_t)b * DIAG_STRIDE;
#pragma unroll
  for (int v = 0; v < 8; ++v) {
    int r = m0 + v + (half << 3);
    float val = -(acc[v] * qmsk[b * LSEQ + r] * cmv);   // dist = -(sim*mask)
    outb[(r + cc) * 128 + r] = val;                     // diagonal-major
  }
}

// ---------------------------------------------------------------------------
// Phase 2: soft-DTW forward value, anti-diagonal wavefront.
// One workgroup (128 threads = 4 wave32) per batch. Triple-buffered diagonals
// in LDS; dist diagonals loaded coalesced from workspace one step ahead.
// softmin(a,b,c) = m - gamma*log(sum exp(-(x-m)/gamma)), gamma = 0.5.
// ---------------------------------------------------------------------------
__global__ __launch_bounds__(128)
void softdtw_diag(const float* __restrict__ wsdiag, float* __restrict__ out) {
  __shared__ float buf[3][132];
  const int b = blockIdx.x;
  const int t = threadIdx.x;
  const float* Dd = wsdiag + (size_t)b * DIAG_STRIDE;

  if (t == 0) {            // R[0][0]=0 ; R[0][1]=R[1][0]=BIG
    buf[0][0] = 0.0f;
    buf[1][0] = BIGV;
    buf[1][1] = BIGV;
  }
  __syncthreads();

  const int i = t + 1;           // this thread owns row i of each diagonal
  float dnext = Dd[t];           // dist diagonal d=2 (slot r = t), prefetched
  for (int d = 2; d <= 256; ++d) {
    const float dc = dnext;
    if (d < 256) dnext = Dd[(d - 1) * 128 + t];   // prefetch next diagonal

    float*       cur = buf[d % 3];
    const float* p1  = buf[(d - 1) % 3];          // diagonal d-1
    const float* p2  = buf[(d + 1) % 3];          // diagonal d-2

    const int j = d - i;
    if (j == 0) {
      cur[i] = BIGV;                              // R[i][0] boundary
    } else if (j >= 1 && j <= 128) {
      float a  = p2[i - 1];                       // R[i-1][j-1]
      float bu = p1[i - 1];                       // R[i-1][j]
      float cl = p1[i];                           // R[i][j-1]
      float m  = fminf(a, fminf(bu, cl));
      float s  = __expf(-2.0f * (a - m)) + __expf(-2.0f * (bu - m)) +
                 __expf(-2.0f * (cl - m));
      float r  = dc + m - 0.5f * __logf(s);
      cur[i] = r;
      if (d == 256 && i == 128) out[b] = -r;      // output = -R[N][M]
    }
    if (t == 0) cur[0] = BIGV;                    // R[0][d] boundary
    __syncthreads();
  }
}

extern "C" void kernel_launch(void* const* d_in, const int* in_sizes, int n_in,
                              void* d_out, int out_size, void* d_ws, size_t ws_size,
                              hipStream_t stream) {
  const float* qry = (const float*)d_in[0];
  const float* cnd = (const float*)d_in[1];
  const float* qm  = (const float*)d_in[2];
  const float* cm  = (const float*)d_in[3];
  float* ws = (float*)d_ws;    // needs 256*32768*4 = 32 MB

  dim3 g1(LSEQ / 16, BATCH);   // (8, 256)
  sim_dist_wmma<<<g1, 256, 0, stream>>>(qry, cnd, qm, cm, ws);
  softdtw_diag<<<BATCH, 128, 0, stream>>>(ws, (float*)d_out);
}